// Attention_56916906606762
// MI455X (gfx1250) — compile-verified
//
#include <hip/hip_runtime.h>
#include <hip/hip_bf16.h>

typedef __attribute__((ext_vector_type(2))) float v2f;
typedef __attribute__((ext_vector_type(8))) float v8f;

static constexpr int NB_HEAD = 20;
static constexpr int DHEAD   = 20;
static constexpr int CIN     = 20;    // input feature dim == NB_HEAD
static constexpr int OUT_DIM = 400;
static constexpr int BATCH   = 32;
static constexpr int LQn     = 512;
static constexpr int LKn     = 512;

__device__ __forceinline__ v8f wmma4(v2f a, v2f b, v8f c) {
  // D = A(16x4 f32) * B(4x16 f32) + C(16x16 f32)
  return __builtin_amdgcn_wmma_f32_16x16x4_f32(
      /*neg_a=*/false, a, /*neg_b=*/false, b,
      /*c_mod=*/(short)0, c, /*reuse_a=*/false, /*reuse_b=*/false);
}

// ---------------------------------------------------------------------------
// Kernel 1: project K_seq and V_seq -> Kp/Vp in [B, H, L, D] layout.
// One wave handles one 16(L) x 16(out-col) tile of the 400-wide projection.
// ---------------------------------------------------------------------------
__global__ __launch_bounds__(32)
void proj_kv(const float* __restrict__ Kseq, const float* __restrict__ Vseq,
             const float* __restrict__ WK,  const float* __restrict__ WKb,
             const float* __restrict__ WV,  const float* __restrict__ WVb,
             float* __restrict__ Kp, float* __restrict__ Vp) {
  const int nt   = blockIdx.x;      // 0..24  (16-col tile of 400)
  const int lt   = blockIdx.y;      // 0..31  (16-row tile of L)
  const int b    = blockIdx.z;
  const int lane = threadIdx.x;
  const int n    = lane & 15;
  const int hi   = lane >> 4;
  const int l0   = lt * 16;
  const int o    = nt * 16 + n;     // output column 0..399 (always valid)

  // A tiles (rows = sequence positions, K-dim = input features), B tiles (W^T)
  v2f ak[5], av[5], bk[5], bv[5];
#pragma unroll
  for (int kc = 0; kc < 5; ++kc) {
    const int c0 = kc * 4 + 2 * hi;
    const float* kr = Kseq + ((size_t)b * LKn + (l0 + n)) * CIN + c0;
    const float* vr = Vseq + ((size_t)b * LKn + (l0 + n)) * CIN + c0;
    ak[kc] = {kr[0], kr[1]};
    av[kc] = {vr[0], vr[1]};
    bk[kc] = {WK[(size_t)o * CIN + c0], WK[(size_t)o * CIN + c0 + 1]};
    bv[kc] = {WV[(size_t)o * CIN + c0], WV[(size_t)o * CIN + c0 + 1]};
  }

  v8f ck = {0.f, 0.f, 0.f, 0.f, 0.f, 0.f, 0.f, 0.f};
  v8f cv = {0.f, 0.f, 0.f, 0.f, 0.f, 0.f, 0.f, 0.f};
#pragma unroll
  for (int kc = 0; kc < 5; ++kc) {
    ck = wmma4(ak[kc], bk[kc], ck);
    cv = wmma4(av[kc], bv[kc], cv);
  }

  const float kb = WKb[o], vb = WVb[o];
  const int h = o / DHEAD, d = o % DHEAD;
#pragma unroll
  for (int i = 0; i < 8; ++i) {
    const int l = l0 + i + 8 * hi;
    const size_t idx = (((size_t)b * NB_HEAD + h) * LKn + l) * DHEAD + d;
    Kp[idx] = ck[i] + kb;
    Vp[idx] = cv[i] + vb;
  }
}

// ---------------------------------------------------------------------------
// Kernel 2: flash attention. One wave per (batch, head, 16-row q-tile).
// Q projection done in-kernel; K/V read from workspace [B,H,L,D].
// ---------------------------------------------------------------------------
__global__ __launch_bounds__(32)
void attn(const float* __restrict__ Qseq,
          const float* __restrict__ WQ, const float* __restrict__ WQb,
          const float* __restrict__ Kp, const float* __restrict__ Vp,
          const int* __restrict__ Qlen, const int* __restrict__ Vlen,
          float* __restrict__ out) {
  __shared__ float ldsQ[16 * 32];   // Q tile staging (C-layout -> A-layout)
  __shared__ float ldsP[16 * 16];   // P tile staging (C-layout -> A-layout)

  const int qt   = blockIdx.x;      // 0..31
  const int h    = blockIdx.y;      // 0..19
  const int b    = blockIdx.z;      // 0..31
  const int lane = threadIdx.x;
  const int n    = lane & 15;
  const int hi   = lane >> 4;
  const int q0   = qt * 16;

  // ---- Q projection for this (b, h, q-tile): X(16x20) * WQ^T slice ----
  v2f ax[5];
#pragma unroll
  for (int kc = 0; kc < 5; ++kc) {
    const int c0 = kc * 4 + 2 * hi;
    const float* xr = Qseq + ((size_t)b * LQn + (q0 + n)) * CIN + c0;
    ax[kc] = {xr[0], xr[1]};
  }
#pragma unroll
  for (int j = 0; j < 2; ++j) {            // two 16-wide N tiles (d 0..15, 16..31)
    const int dcol = j * 16 + n;
    const bool valid = (dcol < DHEAD);
    const int o = h * DHEAD + dcol;
    v8f c = {0.f, 0.f, 0.f, 0.f, 0.f, 0.f, 0.f, 0.f};
#pragma unroll
    for (int kc = 0; kc < 5; ++kc) {
      const int c0 = kc * 4 + 2 * hi;
      v2f bw = {0.f, 0.f};
      if (valid) {
        bw.x = WQ[(size_t)o * CIN + c0];
        bw.y = WQ[(size_t)o * CIN + c0 + 1];
      }
      c = wmma4(ax[kc], bw, c);
    }
    const float bias = valid ? WQb[o] : 0.f;
#pragma unroll
    for (int i = 0; i < 8; ++i)
      ldsQ[(i + 8 * hi) * 32 + j * 16 + n] = c[i] + bias;
  }
  __syncthreads();

  // Reload projected Q in A-layout (M=q row, K=d)
  v2f aq[5];
#pragma unroll
  for (int kc = 0; kc < 5; ++kc) {
    const int c0 = kc * 4 + 2 * hi;
    aq[kc] = {ldsQ[n * 32 + c0], ldsQ[n * 32 + c0 + 1]};
  }

  // ---- flash loop over Lk ----
  const float scale = 0.2236067977499790f;  // 1/sqrt(20)
  const int vlen = Vlen[b];
  const float* Kb_ = Kp + ((size_t)b * NB_HEAD + h) * LKn * DHEAD;
  const float* Vb_ = Vp + ((size_t)b * NB_HEAD + h) * LKn * DHEAD;

  float m[8], lsum[8];
  v8f o0 = {0.f, 0.f, 0.f, 0.f, 0.f, 0.f, 0.f, 0.f};
  v8f o1 = {0.f, 0.f, 0.f, 0.f, 0.f, 0.f, 0.f, 0.f};
#pragma unroll
  for (int i = 0; i < 8; ++i) { m[i] = -1e30f; lsum[i] = 0.f; }

  for (int t = 0; t < LKn / 16; ++t) {
    const int k0 = t * 16;

    // S = Q * K^T : B element (K=d, N=k) = K[k0+n, d]
    v8f s = {0.f, 0.f, 0.f, 0.f, 0.f, 0.f, 0.f, 0.f};
#pragma unroll
    for (int kc = 0; kc < 5; ++kc) {
      const int c0 = kc * 4 + 2 * hi;
      const float* kr = Kb_ + (size_t)(k0 + n) * DHEAD + c0;
      v2f bk = {kr[0], kr[1]};
      s = wmma4(aq[kc], bk, s);
    }

    // scale + additive key mask; online softmax (rows live in 16-lane groups)
    const float kbias = ((k0 + n) >= vlen) ? -1e12f : 0.f;
    float p[8];
#pragma unroll
    for (int i = 0; i < 8; ++i) {
      const float sv = s[i] * scale + kbias;
      float mx = sv;
      mx = fmaxf(mx, __shfl_xor(mx, 1, 32));
      mx = fmaxf(mx, __shfl_xor(mx, 2, 32));
      mx = fmaxf(mx, __shfl_xor(mx, 4, 32));
      mx = fmaxf(mx, __shfl_xor(mx, 8, 32));
      const float mnew = fmaxf(m[i], mx);
      const float pe = __expf(sv - mnew);
      float rs = pe;
      rs += __shfl_xor(rs, 1, 32);
      rs += __shfl_xor(rs, 2, 32);
      rs += __shfl_xor(rs, 4, 32);
      rs += __shfl_xor(rs, 8, 32);
      const float corr = __expf(m[i] - mnew);
      lsum[i] = lsum[i] * corr + rs;
      m[i] = mnew;
      o0[i] *= corr;
      o1[i] *= corr;
      p[i] = pe;
    }

    // Stage P (C-layout) through LDS, reload in A-layout (M=q, K=k)
    __syncthreads();
#pragma unroll
    for (int i = 0; i < 8; ++i) ldsP[(i + 8 * hi) * 16 + n] = p[i];
    __syncthreads();

    v2f ap[4];
#pragma unroll
    for (int kc = 0; kc < 4; ++kc) {
      const int kk = kc * 4 + 2 * hi;
      ap[kc] = {ldsP[n * 16 + kk], ldsP[n * 16 + kk + 1]};
    }

    // O += P * V : B element (K=k row, N=d col) = V[k0+krow, dcol]
#pragma unroll
    for (int kc = 0; kc < 4; ++kc) {
      const int krow = kc * 4 + 2 * hi;
      const float* vr0 = Vb_ + (size_t)(k0 + krow) * DHEAD;
      const float* vr1 = vr0 + DHEAD;
      v2f bv0 = {vr0[n], vr1[n]};
      v2f bv1 = {0.f, 0.f};
      const int dcol = 16 + n;
      if (dcol < DHEAD) { bv1.x = vr0[dcol]; bv1.y = vr1[dcol]; }
      o0 = wmma4(ap[kc], bv0, o0);
      o1 = wmma4(ap[kc], bv1, o1);
    }
  }

  // ---- epilogue: divide by denom, apply q mask, store [B, Lq, 400] ----
  const int qlen = Qlen[b];
#pragma unroll
  for (int i = 0; i < 8; ++i) {
    const int q = q0 + i + 8 * hi;
    const float qm = (q < qlen) ? 1.f : 0.f;
    const float inv = qm / lsum[i];
    const size_t base = ((size_t)b * LQn + q) * OUT_DIM + h * DHEAD;
    out[base + n] = o0[i] * inv;
    if (16 + n < DHEAD) out[base + 16 + n] = o1[i] * inv;
  }
}

// ---------------------------------------------------------------------------
extern "C" void kernel_launch(void* const* d_in, const int* in_sizes, int n_in,
                              void* d_out, int out_size, void* d_ws, size_t ws_size,
                              hipStream_t stream) {
  const float* Qseq = (const float*)d_in[0];
  const float* Kseq = (const float*)d_in[1];
  const float* Vseq = (const float*)d_in[2];
  const int*   Qlen = (const int*)d_in[3];
  const int*   Vlen = (const int*)d_in[4];
  const float* WQ   = (const float*)d_in[5];
  const float* WQb  = (const float*)d_in[6];
  const float* WK   = (const float*)d_in[7];
  const float* WKb  = (const float*)d_in[8];
  const float* WV   = (const float*)d_in[9];
  const float* WVb  = (const float*)d_in[10];
  float* out = (float*)d_out;

  const size_t kvElems = (size_t)BATCH * NB_HEAD * LKn * DHEAD;  // 6.55M floats
  float* Kp = (float*)d_ws;
  float* Vp = Kp + kvElems;   // total ws use: 2 * 26.2 MB = 52.4 MB

  dim3 blk(32);
  dim3 g1(OUT_DIM / 16, LKn / 16, BATCH);   // 25 x 32 x 32 waves
  proj_kv<<<g1, blk, 0, stream>>>(Kseq, Vseq, WK, WKb, WV, WVb, Kp, Vp);

  dim3 g2(LQn / 16, NB_HEAD, BATCH);        // 32 x 20 x 32 waves
  attn<<<g2, blk, 0, stream>>>(Qseq, WQ, WQb, Kp, Vp, Qlen, Vlen, out);
}